// TemplateEmbedder_53300544143981
// MI455X (gfx1250) — compile-verified
//
#include <hip/hip_runtime.h>
#include <hip/hip_bf16.h>
#include <math.h>

// ---------------------------------------------------------------------------
// TemplateEmbedder for MI455X (gfx1250).
//  - fp32 WMMA (V_WMMA_F32_16X16X4_F32) for all contractions (matches fp32 ref,
//    problem is L2-resident / bandwidth bound so lower precision buys nothing).
//  - Block-tiled GEMM: B weight panel staged in LDS by the Tensor Data Mover
//    (tensor_load_to_lds + s_wait_tensorcnt), 4x A-fragment reuse per wave.
//  - Fused flash-style triangle attention (logits never hit HBM).
// ---------------------------------------------------------------------------

#define NTOK   192
#define NPAIR  (NTOK * NTOK)      // 36864
#define C_T    64
#define C_H    32
#define N_HEAD 4
#define C_F    108
#define C_Z    128
#define HID    256                // 4*C_T
#define LN_EPS 1e-5f

typedef __attribute__((ext_vector_type(2))) float v2f;
typedef __attribute__((ext_vector_type(8))) float v8f;
typedef __attribute__((ext_vector_type(4))) unsigned int v4u;
typedef __attribute__((ext_vector_type(8))) int v8i_t;
typedef __attribute__((ext_vector_type(4))) int v4i_t;

__device__ __forceinline__ v8f wmma_f32_16x16x4(v2f a, v2f b, v8f c) {
    return __builtin_amdgcn_wmma_f32_16x16x4_f32(false, a, false, b, (short)0, c,
                                                 false, false);
}

__device__ __forceinline__ float sigmoidf_(float x) {
    return 1.0f / (1.0f + __expf(-x));
}

// ---------------------------------------------------------------------------
// Block-tiled fp32 WMMA GEMM: C[M,N] = A[M,K] * B[K,N], row-major.
// Grid: (N/64, M/128). 256 threads = 8 waves; wave w computes rows
// [tm*128 + w*16, +16) x 64 cols. B panel [K x 64] is DMA'd into LDS by the
// Tensor Data Mover (one descriptor per workgroup), then read via ds_load.
// Requirements: M%128==0, N%64==0, K%4==0, K<=256.
// ---------------------------------------------------------------------------
__global__ __launch_bounds__(256) void gemm_block_kernel(
    const float* __restrict__ A, const float* __restrict__ B,
    float* __restrict__ C, int K, int N)
{
    extern __shared__ float Bs[];              // [K][64]
    const int tn   = blockIdx.x;
    const int tm   = blockIdx.y;
    const int tid  = threadIdx.x;
    const int wave = tid >> 5;
    const int lane = tid & 31;

    // ---- TDM: load B panel (K rows x 64 cols, row stride N) into LDS ----
    if (wave == 0) {
        const unsigned long long gaddr =
            (unsigned long long)(const void*)(B + (size_t)tn * 64);
        const unsigned lds_off = (unsigned)(unsigned long long)(const void*)Bs;

        v4u g0;
        g0[0] = 1u;                                    // count=1, user mode
        g0[1] = lds_off;                               // lds_addr (bytes)
        g0[2] = (unsigned)(gaddr & 0xFFFFFFFFu);       // global_addr[31:0]
        g0[3] = (unsigned)((gaddr >> 32) & 0x01FFFFFFu)// global_addr[56:32]
              | (2u << 30);                            // type = 2 ("image")

        v8i_t g1;
        g1[0] = 0x20000;                 // wg_mask=0, data_size=2 (4 bytes)
        g1[1] = (64 << 16);              // tensor_dim0[15:0] = 64
        g1[2] = (K & 0xFFFF) << 16;      // tensor_dim0 hi=0 | tensor_dim1 lo=K
        g1[3] = (64 << 16);              // tensor_dim1 hi=0 | tile_dim0 = 64
        g1[4] = K;                       // tile_dim1 = K, tile_dim2 = 0
        g1[5] = N;                       // tensor_dim0_stride[31:0] = N
        g1[6] = 0;                       // stride0 hi | stride1 lo
        g1[7] = 0;                       // stride1 hi

        v4i_t gz4 = {0, 0, 0, 0};        // groups 2/3 unused (2-D tensor)
        v8i_t gz8 = {0, 0, 0, 0, 0, 0, 0, 0};
        // 6-arg toolchain form: (g0, g1, g2, g3, g4, cpol)
        __builtin_amdgcn_tensor_load_to_lds(g0, g1, gz4, gz4, gz8, 0);
        __builtin_amdgcn_s_wait_tensorcnt((short)0);
    }
    __syncthreads();

    const int r  = lane & 15;
    const int ks = (lane >> 4) * 2;
    const float* Ap = A + (size_t)(tm * 128 + wave * 16 + r) * K;
    __builtin_prefetch(Ap, 0, 3);        // global_prefetch_b8 of this A row

    v8f acc0 = {}, acc1 = {}, acc2 = {}, acc3 = {};
    for (int k = 0; k < K; k += 4) {
        v2f a;
        a.x = Ap[k + ks];
        a.y = Ap[k + ks + 1];
        const float* b0 = Bs + (size_t)(k + ks) * 64;
        const float* b1 = Bs + (size_t)(k + ks + 1) * 64;
        v2f b;
        b.x = b0[r];      b.y = b1[r];      acc0 = wmma_f32_16x16x4(a, b, acc0);
        b.x = b0[16 + r]; b.y = b1[16 + r]; acc1 = wmma_f32_16x16x4(a, b, acc1);
        b.x = b0[32 + r]; b.y = b1[32 + r]; acc2 = wmma_f32_16x16x4(a, b, acc2);
        b.x = b0[48 + r]; b.y = b1[48 + r]; acc3 = wmma_f32_16x16x4(a, b, acc3);
    }

    const int col0 = tn * 64 + (lane & 15);
    const int rb   = tm * 128 + wave * 16 + ((lane >> 4) << 3);
    for (int i = 0; i < 8; ++i) {
        float* Cr = C + (size_t)(rb + i) * N + col0;
        Cr[0]  = acc0[i];
        Cr[16] = acc1[i];
        Cr[32] = acc2[i];
        Cr[48] = acc3[i];
    }
}

// ---------------------------------------------------------------------------
// Triangle-mul einsum: 32 channel-batched 192x192x192 fp32 GEMMs.
//   outgoing: x[i,j,c] = sum_k a[i,k,c] * b[j,k,c]   (A * B^T)
//   incoming: x[i,j,c] = sum_k a[k,i,c] * b[k,j,c]   (A^T * B)
// ab layout [N,N,64]; a = channels 0..31, b = channels 32..63. x: [N,N,32].
// ---------------------------------------------------------------------------
__global__ __launch_bounds__(32) void trimul_einsum_kernel(
    const float* __restrict__ ab, float* __restrict__ x, int outgoing)
{
    const int c  = blockIdx.z;       // 0..31
    const int tj = blockIdx.x;       // 0..11
    const int ti = blockIdx.y;       // 0..11
    const float* a = ab + c;
    const float* b = ab + 32 + c;

    const int lane = threadIdx.x;
    const int r  = lane & 15;
    const int ks = (lane >> 4) * 2;
    const int i0 = ti * 16, j0 = tj * 16;

    v8f acc = {};
    if (outgoing) {
        for (int k = 0; k < NTOK; k += 4) {
            v2f av, bv;
            av.x = a[((size_t)(i0 + r) * NTOK + k + ks) * 64];
            av.y = a[((size_t)(i0 + r) * NTOK + k + ks + 1) * 64];
            bv.x = b[((size_t)(j0 + r) * NTOK + k + ks) * 64];
            bv.y = b[((size_t)(j0 + r) * NTOK + k + ks + 1) * 64];
            acc = wmma_f32_16x16x4(av, bv, acc);
        }
    } else {
        for (int k = 0; k < NTOK; k += 4) {
            v2f av, bv;
            av.x = a[((size_t)(k + ks) * NTOK + i0 + r) * 64];
            av.y = a[((size_t)(k + ks + 1) * NTOK + i0 + r) * 64];
            bv.x = b[((size_t)(k + ks) * NTOK + j0 + r) * 64];
            bv.y = b[((size_t)(k + ks + 1) * NTOK + j0 + r) * 64];
            acc = wmma_f32_16x16x4(av, bv, acc);
        }
    }
    const int col = j0 + (lane & 15);
    const int rb  = i0 + ((lane >> 4) << 3);
    for (int i = 0; i < 8; ++i)
        x[((size_t)(rb + i) * NTOK + col) * 32 + c] = acc[i];
}

// ---------------------------------------------------------------------------
// LayerNorm: one wave per row, C in {32,64,128}.
// ---------------------------------------------------------------------------
__global__ __launch_bounds__(256) void ln_kernel(
    const float* __restrict__ x, const float* __restrict__ g,
    const float* __restrict__ b, float* __restrict__ y, int C, int rows)
{
    const int wave = blockIdx.x * (blockDim.x >> 5) + (threadIdx.x >> 5);
    const int lane = threadIdx.x & 31;
    if (wave >= rows) return;

    const float* xr = x + (size_t)wave * C;
    float* yr = y + (size_t)wave * C;
    const int per = C >> 5;              // 1, 2 or 4
    float vals[4];
    float s = 0.f;
    for (int i = 0; i < per; ++i) { vals[i] = xr[lane + i * 32]; s += vals[i]; }
    for (int off = 16; off; off >>= 1) s += __shfl_xor(s, off, 32);
    const float mu = s / (float)C;
    float q = 0.f;
    for (int i = 0; i < per; ++i) { float d = vals[i] - mu; q += d * d; }
    for (int off = 16; off; off >>= 1) q += __shfl_xor(q, off, 32);
    const float rs = rsqrtf(q / (float)C + LN_EPS);
    for (int i = 0; i < per; ++i) {
        const int c = lane + i * 32;
        yr[c] = (vals[i] - mu) * rs * g[c] + b[c];
    }
}

// ---------------------------------------------------------------------------
// Elementwise helpers
// ---------------------------------------------------------------------------
__global__ void ew_zero_kernel(float* __restrict__ p, size_t n) {
    size_t i = (size_t)blockIdx.x * blockDim.x + threadIdx.x;
    if (i < n) p[i] = 0.f;
}
__global__ void ew_add_kernel(float* __restrict__ d, const float* __restrict__ s, size_t n) {
    size_t i = (size_t)blockIdx.x * blockDim.x + threadIdx.x;
    if (i < n) d[i] += s[i];
}
__global__ void ew_sigmul_kernel(const float* __restrict__ s1, const float* __restrict__ s2,
                                 float* __restrict__ o, size_t n) {
    size_t i = (size_t)blockIdx.x * blockDim.x + threadIdx.x;
    if (i < n) o[i] = sigmoidf_(s1[i]) * s2[i];
}
__global__ void ew_silumul_kernel(const float* __restrict__ s1, const float* __restrict__ s2,
                                  float* __restrict__ o, size_t n) {
    size_t i = (size_t)blockIdx.x * blockDim.x + threadIdx.x;
    if (i < n) { float a = s1[i]; o[i] = a * sigmoidf_(a) * s2[i]; }
}
__global__ void ew_gate_add_kernel(float* __restrict__ v, const float* __restrict__ graw,
                                   const float* __restrict__ y, size_t n) {
    size_t i = (size_t)blockIdx.x * blockDim.x + threadIdx.x;
    if (i < n) v[i] += sigmoidf_(graw[i]) * y[i];
}
__global__ void ew_relu_scale_kernel(const float* __restrict__ u, float* __restrict__ o,
                                     float sc, size_t n) {
    size_t i = (size_t)blockIdx.x * blockDim.x + threadIdx.x;
    if (i < n) o[i] = fmaxf(u[i] * sc, 0.f);
}

// bias = zn[M,64] @ Wb[64,4]  (N=4 too narrow for WMMA; tiny, VALU is fine)
__global__ void bias_proj_kernel(const float* __restrict__ zn, const float* __restrict__ Wb,
                                 float* __restrict__ bias, int M) {
    int m = blockIdx.x * blockDim.x + threadIdx.x;
    if (m >= M) return;
    const float* z = zn + (size_t)m * 64;
    float acc[4] = {0.f, 0.f, 0.f, 0.f};
    for (int c = 0; c < 64; ++c) {
        float zv = z[c];
        for (int h = 0; h < 4; ++h) acc[h] += zv * Wb[c * 4 + h];
    }
    for (int h = 0; h < 4; ++h) bias[(size_t)m * 4 + h] = acc[h];
}

// ---------------------------------------------------------------------------
// Fused triangle attention: one workgroup per (i, head). K/V tiles staged in
// LDS, logits computed with WMMA into per-wave LDS scratch, row softmax, then
// probs @ V with WMMA. Never materializes the [h,i,j,k] logits in HBM.
// Dynamic LDS: 192*32*2 (K,V) + 192 (bias col) + 8*16*192 (logits) floats.
// ---------------------------------------------------------------------------
__global__ __launch_bounds__(256) void attn_kernel(
    const float* __restrict__ qkv,   // [NPAIR, 384]  (3, head, c) innermost
    const float* __restrict__ bias,  // [NPAIR, 4]
    const float* __restrict__ graw,  // [NPAIR, 128]  pre-sigmoid gate
    float* __restrict__ og,          // [NPAIR, 128]  gated attention out
    int starting)
{
    extern __shared__ float smem[];
    float* Ks    = smem;                    // [192][32]
    float* Vs    = Ks + NTOK * 32;          // [192][32]
    float* biasK = Vs + NTOK * 32;          // [192] (ending variant)
    float* Ls    = biasK + NTOK;            // [8][16][192]

    const int i = blockIdx.x;
    const int h = blockIdx.y;
    const int tid  = threadIdx.x;
    const int wave = tid >> 5;
    const int lane = tid & 31;

    for (int idx = tid; idx < NTOK * 32; idx += 256) {
        const int kk = idx >> 5, c = idx & 31;
        const size_t src = starting ? (size_t)(i * NTOK + kk) * 384
                                    : (size_t)(kk * NTOK + i) * 384;
        Ks[kk * 32 + c] = qkv[src + 128 + h * 32 + c];
        Vs[kk * 32 + c] = qkv[src + 256 + h * 32 + c];
    }
    if (!starting) {
        for (int kk = tid; kk < NTOK; kk += 256)
            biasK[kk] = bias[(size_t)(kk * NTOK + i) * 4 + h];
    }
    __syncthreads();

    const float scale = 0.17677669529663687f;  // 1/sqrt(32)
    const int r  = lane & 15;
    const int ks = (lane >> 4) * 2;
    float* Lw = Ls + wave * 16 * NTOK;

    for (int jt = wave; jt < NTOK / 16; jt += 8) {
        const int j0 = jt * 16;
        const float* Qb = qkv + (size_t)(i * NTOK + j0 + r) * 384 + h * 32;

        // logits[16 x 192] = Q (16x32) * K^T
        for (int kt = 0; kt < NTOK / 16; ++kt) {
            v8f acc = {};
            for (int c = 0; c < 32; c += 4) {
                v2f a, b;
                a.x = Qb[c + ks];
                a.y = Qb[c + ks + 1];
                b.x = Ks[(kt * 16 + r) * 32 + c + ks];
                b.y = Ks[(kt * 16 + r) * 32 + c + ks + 1];
                acc = wmma_f32_16x16x4(a, b, acc);
            }
            const int col = kt * 16 + (lane & 15);
            const int rb  = (lane >> 4) << 3;
            for (int q8 = 0; q8 < 8; ++q8)
                Lw[(rb + q8) * NTOK + col] = acc[q8];
        }

        // softmax over k (192) for each of 16 rows, wave-cooperative
        for (int jr = 0; jr < 16; ++jr) {
            const int j = j0 + jr;
            float e[6];
            float m = -3.4e38f;
            for (int t = 0; t < 6; ++t) {
                const int kk = lane + t * 32;
                float l = Lw[jr * NTOK + kk] * scale;
                l += starting ? bias[(size_t)(j * NTOK + kk) * 4 + h] : biasK[kk];
                e[t] = l;
                m = fmaxf(m, l);
            }
            for (int off = 16; off; off >>= 1) m = fmaxf(m, __shfl_xor(m, off, 32));
            float s = 0.f;
            for (int t = 0; t < 6; ++t) { e[t] = __expf(e[t] - m); s += e[t]; }
            for (int off = 16; off; off >>= 1) s += __shfl_xor(s, off, 32);
            const float inv = 1.f / s;
            for (int t = 0; t < 6; ++t) Lw[jr * NTOK + lane + t * 32] = e[t] * inv;
        }

        // O[16 x 32] = probs (16x192) * V (192x32), then gate & store
        for (int ct = 0; ct < 2; ++ct) {
            v8f acc = {};
            for (int kk = 0; kk < NTOK; kk += 4) {
                v2f a, b;
                a.x = Lw[r * NTOK + kk + ks];
                a.y = Lw[r * NTOK + kk + ks + 1];
                b.x = Vs[(kk + ks) * 32 + ct * 16 + r];
                b.y = Vs[(kk + ks + 1) * 32 + ct * 16 + r];
                acc = wmma_f32_16x16x4(a, b, acc);
            }
            const int col = ct * 16 + (lane & 15);
            const int rb  = (lane >> 4) << 3;
            for (int q8 = 0; q8 < 8; ++q8) {
                const int j = j0 + rb + q8;
                const size_t o = (size_t)(i * NTOK + j) * 128 + h * 32 + col;
                og[o] = sigmoidf_(graw[o]) * acc[q8];
            }
        }
    }
}

// ===========================================================================
// Host side
// ===========================================================================
namespace {

struct TriMulP  { const float *W1, *W2, *W3, *W4, *ln_b, *ln_g, *ln4_b, *ln4_g; };
struct TriAttnP { const float *Wb, *Wg, *Wo, *Wqkv, *ln_b, *ln_g; };
struct TransP   { const float *W1, *W2, *W3, *ln_b, *ln_g; };
struct BlockP   { TriAttnP tae, tas; TriMulP tmi, tmo; TransP tr; };

struct Workspace {
    float *zp, *v, *u, *zn, *t0, *t1;
};

inline void launch_gemm(const float* A, const float* B, float* C,
                        int M, int N, int K, hipStream_t s) {
    dim3 grid(N / 64, M / 128);
    size_t smem = (size_t)K * 64 * sizeof(float);
    gemm_block_kernel<<<grid, 256, smem, s>>>(A, B, C, K, N);
}
inline void launch_ln(const float* x, const float* g, const float* b, float* y,
                      int C, int rows, hipStream_t s) {
    int grid = (rows + 7) / 8;
    ln_kernel<<<grid, 256, 0, s>>>(x, g, b, y, C, rows);
}
inline dim3 ew_grid(size_t n) { return dim3((unsigned)((n + 255) / 256)); }

void tri_mul(float* v, const TriMulP& p, int outgoing, const Workspace& w, hipStream_t s) {
    const size_t NP = NPAIR;
    float* s1   = w.t0;
    float* s2   = w.t0 + NP * 64;
    float* graw = w.t0 + 2 * NP * 64;
    float* x    = w.t0 + 3 * NP * 64;        // NP*32
    float* xn   = x + NP * 32;               // NP*32
    float* y    = w.t1;                      // NP*64

    launch_ln(v, p.ln_g, p.ln_b, w.zn, 64, NP, s);
    launch_gemm(w.zn, p.W1, s1, NP, 64, 64, s);
    launch_gemm(w.zn, p.W2, s2, NP, 64, 64, s);
    ew_sigmul_kernel<<<ew_grid(NP * 64), 256, 0, s>>>(s1, s2, s1, NP * 64);  // ab
    launch_gemm(w.zn, p.W3, graw, NP, 64, 64, s);
    trimul_einsum_kernel<<<dim3(12, 12, 32), 32, 0, s>>>(s1, x, outgoing);
    launch_ln(x, p.ln4_g, p.ln4_b, xn, 32, NP, s);
    launch_gemm(xn, p.W4, y, NP, 64, 32, s);
    ew_gate_add_kernel<<<ew_grid(NP * 64), 256, 0, s>>>(v, graw, y, NP * 64);
}

void tri_attn(float* v, const TriAttnP& p, int starting, const Workspace& w, hipStream_t s) {
    const size_t NP = NPAIR;
    float* qkv  = w.t0;                  // NP*384
    float* bias = w.t1;                  // NP*4
    float* graw = w.t1 + NP * 4;         // NP*128
    float* og   = w.t1 + NP * 132;       // NP*128
    float* y    = w.t0;                  // reuse after attn consumes qkv

    launch_ln(v, p.ln_g, p.ln_b, w.zn, 64, NP, s);
    launch_gemm(w.zn, p.Wqkv, qkv, NP, 384, 64, s);
    bias_proj_kernel<<<ew_grid(NP), 256, 0, s>>>(w.zn, p.Wb, bias, NP);
    launch_gemm(w.zn, p.Wg, graw, NP, 128, 64, s);

    const size_t smem = (size_t)(NTOK * 32 * 2 + NTOK + 8 * 16 * NTOK) * sizeof(float);
    attn_kernel<<<dim3(NTOK, N_HEAD), 256, smem, s>>>(qkv, bias, graw, og, starting);

    launch_gemm(og, p.Wo, y, NP, 64, 128, s);
    ew_add_kernel<<<ew_grid(NP * 64), 256, 0, s>>>(v, y, NP * 64);
}

void transition(float* v, const TransP& p, const Workspace& w, hipStream_t s) {
    const size_t NP = NPAIR;
    float* h1 = w.t0;    // NP*256
    float* h2 = w.t1;    // NP*256

    launch_ln(v, p.ln_g, p.ln_b, w.zn, 64, NP, s);
    launch_gemm(w.zn, p.W1, h1, NP, HID, 64, s);
    launch_gemm(w.zn, p.W2, h2, NP, HID, 64, s);
    ew_silumul_kernel<<<ew_grid(NP * HID), 256, 0, s>>>(h1, h2, h1, NP * HID);
    launch_gemm(h1, p.W3, h2, NP, 64, HID, s);
    ew_add_kernel<<<ew_grid(NP * 64), 256, 0, s>>>(v, h2, NP * 64);
}

} // namespace

extern "C" void kernel_launch(void* const* d_in, const int* in_sizes, int n_in,
                              void* d_out, int out_size, void* d_ws, size_t ws_size,
                              hipStream_t stream) {
    (void)in_sizes; (void)n_in; (void)out_size; (void)ws_size;
    const size_t NP = NPAIR;

    // ---- inputs (pytree flatten order: dict keys sorted, lists in order) ----
    const float* f    = (const float*)d_in[0];   // [1,2,192,192,108]
    const float* z_ij = (const float*)d_in[1];   // [1,192,192,128]
    int p = 2;
    auto nxt = [&]() { return (const float*)d_in[p++]; };

    BlockP blocks[2];
    for (int b = 0; b < 2; ++b) {
        BlockP& B = blocks[b];
        // block dict sorted: tae, tas, tmi, tmo, tr
        // triattn sorted: Wb, Wg, Wo, Wqkv, ln(b,g)
        B.tae = { nxt(), nxt(), nxt(), nxt(), nxt(), nxt() };
        B.tas = { nxt(), nxt(), nxt(), nxt(), nxt(), nxt() };
        // trimul sorted: W1, W2, W3, W4, ln(b,g), ln4(b,g)
        B.tmi = { nxt(), nxt(), nxt(), nxt(), nxt(), nxt(), nxt(), nxt() };
        B.tmo = { nxt(), nxt(), nxt(), nxt(), nxt(), nxt(), nxt(), nxt() };
        // trans sorted: W1, W2, W3, ln(b,g)
        B.tr  = { nxt(), nxt(), nxt(), nxt(), nxt() };
    }
    // outer sorted: Wf, Wout, Wz, ln_v(b,g), ln_z(b,g)
    const float* Wf    = nxt();
    const float* Wout  = nxt();
    const float* Wz    = nxt();
    const float* lnv_b = nxt();
    const float* lnv_g = nxt();
    const float* lnz_b = nxt();
    const float* lnz_g = nxt();

    // ---- workspace carve (floats): 4*NP*64 + NP*384 + NP*260 = NP*900 ----
    Workspace w;
    float* ws = (float*)d_ws;
    w.zp = ws;
    w.v  = w.zp + NP * 64;
    w.u  = w.v + NP * 64;
    w.zn = w.u + NP * 64;
    w.t0 = w.zn + NP * 64;           // NP*384
    w.t1 = w.t0 + NP * 384;          // NP*260

    // ---- zp = LN(z_ij) @ Wz ----
    launch_ln(z_ij, lnz_g, lnz_b, w.t0, C_Z, NP, stream);          // t0: [NP,128]
    launch_gemm(w.t0, Wz, w.zp, NP, C_T, C_Z, stream);

    ew_zero_kernel<<<ew_grid(NP * 64), 256, 0, stream>>>(w.u, NP * 64);

    for (int t = 0; t < 2; ++t) {
        // v = f[:,t] @ Wf + zp
        const float* ft = f + (size_t)t * NP * C_F;
        launch_gemm(ft, Wf, w.v, NP, C_T, C_F, stream);
        ew_add_kernel<<<ew_grid(NP * 64), 256, 0, stream>>>(w.v, w.zp, NP * 64);

        for (int b = 0; b < 2; ++b) {
            tri_mul (w.v, blocks[b].tmo, /*outgoing=*/1, w, stream);
            tri_mul (w.v, blocks[b].tmi, /*outgoing=*/0, w, stream);
            tri_attn(w.v, blocks[b].tas, /*starting=*/1, w, stream);
            tri_attn(w.v, blocks[b].tae, /*starting=*/0, w, stream);
            transition(w.v, blocks[b].tr, w, stream);
        }
        // u += LN(v)
        launch_ln(w.v, lnv_g, lnv_b, w.zn, 64, NP, stream);
        ew_add_kernel<<<ew_grid(NP * 64), 256, 0, stream>>>(w.u, w.zn, NP * 64);
    }

    // out = relu(u / n_templ) @ Wout
    ew_relu_scale_kernel<<<ew_grid(NP * 64), 256, 0, stream>>>(w.u, w.zn, 0.5f, NP * 64);
    launch_gemm(w.zn, Wout, (float*)d_out, NP, C_Z, C_T, stream);
}